// FCOSDetector_53128745452217
// MI455X (gfx1250) — compile-verified
//
#include <hip/hip_runtime.h>
#include <hip/hip_bf16.h>
#include <math.h>

// ---------------------------------------------------------------------------
// FCOS detector post-process for MI455X (gfx1250, wave32)
//   stage 1: decode   (streaming, ~87MB read -> HBM-bound, ~4us ideal)
//   stage 2: top-k    (per-batch 8-bit radix select + 1024 bitonic sort, LDS)
//   stage 3: NMS+out  (per-batch sequential greedy NMS in LDS, exact ref math)
// Box decode D = W(4x6)*(x,y,l,t,r,b) is done with V_WMMA_F32_16X16X4_F32
// (two chained K=4 WMMAs per 16 locations), f32-exact.
// ---------------------------------------------------------------------------

#define NB    16
#define NLOC  17064
#define NCLS  80
#define KTOP  1000
#define IMG_W 1024
#define IMG_H 800

typedef __attribute__((ext_vector_type(2))) float v2f;
typedef __attribute__((ext_vector_type(8))) float v8f;

#ifndef __has_builtin
#define __has_builtin(x) 0
#endif
#if __has_builtin(__builtin_amdgcn_wmma_f32_16x16x4_f32)
#define USE_WMMA_DECODE 1
#else
#define USE_WMMA_DECODE 0
#endif

struct LevelPtrs {
    const float* cls[5];
    const float* cnt[5];
    const float* reg[5];
};

__device__ __forceinline__ float sigmoidf_(float x) {
    return 1.0f / (1.0f + expf(-x));
}

// ---------------------------------------------------------------------------
// Stage 1: per-location score / class / box
// ---------------------------------------------------------------------------
__global__ __launch_bounds__(256)
void decode_kernel(LevelPtrs P, float* ws_scores, int* ws_cls, float* ws_boxes)
{
    const int lvl_off[6] = {0, 12800, 16000, 16800, 17008, 17064};
    const int lvl_w[5]   = {128, 64, 32, 16, 8};
    const int lvl_s[5]   = {8, 16, 32, 64, 128};

    const int total = NB * NLOC;
    const int gid   = blockIdx.x * 256 + threadIdx.x;
    const int idx   = (gid < total) ? gid : (total - 1);  // clamp tail, keep EXEC full
    const int b     = idx / NLOC;
    const int n     = idx - b * NLOC;

    int lvl = 0;
    while (lvl < 4 && n >= lvl_off[lvl + 1]) lvl++;
    const int rem = n - lvl_off[lvl];
    const int w   = lvl_w[lvl];
    const int s   = lvl_s[lvl];
    const int hw  = lvl_off[lvl + 1] - lvl_off[lvl];
    const int iy  = rem / w;
    const int ix  = rem - iy * w;
    const float cx = (float)(ix * s + (s >> 1));
    const float cy = (float)(iy * s + (s >> 1));

    // class max/argmax over 80 channels (stride hw); sigmoid is monotone so
    // max(sigmoid) == sigmoid(max-logit) and argmax is identical.
    const float* cls = P.cls[lvl] + (size_t)b * NCLS * hw + rem;
    float best = cls[0];
    int bestc  = 0;
    for (int c = 1; c < NCLS; c++) {
        if (c + 8 < NCLS) __builtin_prefetch(cls + (size_t)(c + 8) * hw, 0, 1);
        float v = cls[(size_t)c * hw];
        if (v > best) { best = v; bestc = c; }
    }

    const float cntl = P.cnt[lvl][(size_t)b * hw + rem];
    const float sc   = sqrtf(sigmoidf_(best) * sigmoidf_(cntl));

    const float* reg = P.reg[lvl] + (size_t)b * 4 * hw + rem;
    const float l  = reg[0];
    const float t  = reg[(size_t)hw];
    const float r  = reg[(size_t)2 * hw];
    const float bt = reg[(size_t)3 * hw];

    if (gid < total) {
        ws_scores[idx] = sc;
        ws_cls[idx]    = bestc + 1;
    }

#if USE_WMMA_DECODE
    // box = W(4x6)*(x,y,l,t,r,b), split into two K=4 WMMAs:
    //   pass1: K rows = (x,y,l,t): W1 = [[1,0,-1,0],[0,1,0,-1],[1,0,0,0],[0,1,0,0]]
    //   pass2: K rows = (r,b,0,0): W2 = [[0,0,0,0],[0,0,0,0],[1,0,0,0],[0,1,0,0]]
    // A 16x4 f32 layout: lanes0-15=M, V0={K0|K2}, V1={K1|K3}. B 4x16 mirrored.
    // D 16x16 f32: VGPR v, lanes0-15 -> M=v, N=lane.
    const int lane     = threadIdx.x & 31;
    const int waveBase = (blockIdx.x * 256 + threadIdx.x) - lane;
    const bool lo      = lane < 16;
    const int  m       = lane & 15;

    v2f A1, A2;
    A1[0] = lo ? ((m == 0 || m == 2) ? 1.f : 0.f) : ((m == 0) ? -1.f : 0.f);
    A1[1] = lo ? ((m == 1 || m == 3) ? 1.f : 0.f) : ((m == 1) ? -1.f : 0.f);
    A2[0] = lo ? ((m == 2) ? 1.f : 0.f) : 0.f;
    A2[1] = lo ? ((m == 3) ? 1.f : 0.f) : 0.f;

    for (int g = 0; g < 2; g++) {
        const int srcl = g * 16 + m;   // column owner lane within wave
        float sx = __shfl(cx, srcl, 32), sy = __shfl(cy, srcl, 32);
        float sl = __shfl(l,  srcl, 32), st = __shfl(t,  srcl, 32);
        float sr = __shfl(r,  srcl, 32), sb = __shfl(bt, srcl, 32);

        v2f B1, B2;
        B1[0] = lo ? sx : sl;   // K0 | K2
        B1[1] = lo ? sy : st;   // K1 | K3
        B2[0] = lo ? sr : 0.f;
        B2[1] = lo ? sb : 0.f;

        v8f acc = {};
        acc = __builtin_amdgcn_wmma_f32_16x16x4_f32(false, A1, false, B1,
                                                    (short)0, acc, false, false);
        acc = __builtin_amdgcn_wmma_f32_16x16x4_f32(false, A2, false, B2,
                                                    (short)0, acc, false, false);

        const int col_gid = waveBase + g * 16 + lane;  // valid for lanes 0-15
        if (lo && col_gid < total) {
            float* bo = ws_boxes + (size_t)col_gid * 4;
            bo[0] = acc[0]; bo[1] = acc[1]; bo[2] = acc[2]; bo[3] = acc[3];
        }
    }
#else
    if (gid < total) {
        float* bo = ws_boxes + (size_t)idx * 4;
        bo[0] = cx - l; bo[1] = cy - t; bo[2] = cx + r; bo[3] = cy + bt;
    }
#endif
}

// ---------------------------------------------------------------------------
// Stage 2: per-batch sorted top-1000 (radix select -> ordered compaction ->
// bitonic sort of 1024 in LDS). Ties broken by ascending index (jax top_k).
// ---------------------------------------------------------------------------
__global__ __launch_bounds__(1024)
void topk_kernel(const float* ws_scores, int* top_idx, float* top_scores)
{
    __shared__ int      hist[256];
    __shared__ int      scan[1024];
    __shared__ unsigned selK[1024];
    __shared__ int      selI[1024];
    __shared__ unsigned sh_prefix;
    __shared__ int      sh_remaining;

    const int b   = blockIdx.x;
    const int tid = threadIdx.x;
    const float* sc = ws_scores + (size_t)b * NLOC;

    // scores are >= 0 so key = bits | 0x80000000 is order-preserving
    unsigned prefix = 0;
    int remaining   = KTOP;
    for (int pass = 3; pass >= 0; pass--) {
        if (tid < 256) hist[tid] = 0;
        __syncthreads();
        const int shift = pass * 8;
        for (int n = tid; n < NLOC; n += 1024) {
            unsigned key = __float_as_uint(sc[n]) | 0x80000000u;
            if (pass == 3 || (key >> (shift + 8)) == prefix)
                atomicAdd(&hist[(key >> shift) & 0xFF], 1);
        }
        __syncthreads();
        if (tid == 0) {
            int rem = remaining;
            unsigned bin = 0;
            for (int v = 255; v >= 0; v--) {
                if (hist[v] >= rem) { bin = (unsigned)v; break; }
                rem -= hist[v];
            }
            sh_prefix    = (prefix << 8) | bin;
            sh_remaining = rem;
        }
        __syncthreads();
        prefix    = sh_prefix;
        remaining = sh_remaining;
        __syncthreads();
    }
    const unsigned T = prefix;        // exact key of the KTOP-th largest
    const int G = KTOP - remaining;   // count of keys strictly greater than T

    selK[tid] = 0u;
    selI[tid] = 0x7FFFFFFF;
    __syncthreads();

    // ordered compaction: all key>T, then first `remaining` of key==T by index
    int baseGt = 0, baseEq = 0;
    for (int start = 0; start < NLOC; start += 1024) {
        const int n = start + tid;
        unsigned key = 0;
        if (n < NLOC) key = __float_as_uint(sc[n]) | 0x80000000u;
        const int pgt = (n < NLOC && key >  T) ? 1 : 0;
        const int peq = (n < NLOC && key == T) ? 1 : 0;
        const int v   = pgt | (peq << 16);
        scan[tid] = v;
        __syncthreads();
        for (int off = 1; off < 1024; off <<= 1) {
            int add = (tid >= off) ? scan[tid - off] : 0;
            __syncthreads();
            scan[tid] += add;
            __syncthreads();
        }
        const int excl = scan[tid] - v;
        const int tot  = scan[1023];
        if (pgt) {
            int slot = baseGt + (excl & 0xFFFF);
            selK[slot] = key; selI[slot] = n;
        }
        if (peq) {
            int e = baseEq + (excl >> 16);
            if (e < remaining) { selK[G + e] = key; selI[G + e] = n; }
        }
        baseGt += tot & 0xFFFF;
        baseEq += tot >> 16;
        __syncthreads();
    }

    // bitonic sort 1024: descending key, ascending index on ties
    for (int k = 2; k <= 1024; k <<= 1) {
        for (int j = k >> 1; j > 0; j >>= 1) {
            __syncthreads();
            const int i = tid, ixj = i ^ j;
            if (ixj > i) {
                unsigned ka = selK[i], kb = selK[ixj];
                int      ia = selI[i], ib = selI[ixj];
                const bool up  = ((i & k) == 0);
                const bool aft = (ka < kb) || (ka == kb && ia > ib); // a after b
                const bool bef = (ka > kb) || (ka == kb && ia < ib); // a before b
                if (up ? aft : bef) {
                    selK[i] = kb; selK[ixj] = ka;
                    selI[i] = ib; selI[ixj] = ia;
                }
            }
        }
    }
    __syncthreads();
    if (tid < KTOP) {
        top_idx[b * KTOP + tid]    = selI[tid];
        top_scores[b * KTOP + tid] = __uint_as_float(selK[tid] ^ 0x80000000u);
    }
}

// ---------------------------------------------------------------------------
// Stage 3: class-offset greedy NMS (exact reference recurrence) + outputs
// d_out layout (floats): scores[16000] | classes[16000] | boxes[64000] | keep[16000]
// ---------------------------------------------------------------------------
__global__ __launch_bounds__(1024)
void nms_kernel(const int* ws_cls, const float* ws_boxes,
                const int* top_idx, const float* top_scores, float* out)
{
    __shared__ float X1[KTOP], Y1[KTOP], X2[KTOP], Y2[KTOP], AR[KTOP];
    __shared__ int   VAL[KTOP], SUP[KTOP], KEEP[KTOP];
    __shared__ float red[1024];

    const int b   = blockIdx.x;
    const int tid = threadIdx.x;

    float rx1 = 0.f, ry1 = 0.f, rx2 = 0.f, ry2 = 0.f, scv = 0.f;
    int clsv = 0;
    float localMax = -3.4e38f;
    if (tid < KTOP) {
        const int idx  = top_idx[b * KTOP + tid];
        const float* bo = ws_boxes + ((size_t)b * NLOC + idx) * 4;
        rx1 = bo[0]; ry1 = bo[1]; rx2 = bo[2]; ry2 = bo[3];
        scv  = top_scores[b * KTOP + tid];
        clsv = ws_cls[(size_t)b * NLOC + idx];
        localMax = fmaxf(fmaxf(rx1, ry1), fmaxf(rx2, ry2));
    }
    red[tid] = localMax;
    __syncthreads();
    for (int off = 512; off > 0; off >>= 1) {
        if (tid < off) red[tid] = fmaxf(red[tid], red[tid + off]);
        __syncthreads();
    }
    const float maxc = red[0];

    if (tid < KTOP) {
        const float o = (float)clsv * (maxc + 1.0f);
        X1[tid] = rx1 + o; Y1[tid] = ry1 + o;
        X2[tid] = rx2 + o; Y2[tid] = ry2 + o;
        AR[tid] = (X2[tid] - X1[tid] + 1.f) * (Y2[tid] - Y1[tid] + 1.f);
        VAL[tid] = (scv >= 0.05f) ? 1 : 0;
        SUP[tid] = 0;
        KEEP[tid] = 0;
    }
    __syncthreads();

    for (int i = 0; i < KTOP; i++) {
        const bool keep_i = (VAL[i] != 0) && (SUP[i] == 0);
        if (tid == i) KEEP[i] = keep_i ? 1 : 0;
        if (keep_i && tid > i && tid < KTOP) {
            float xmin = fmaxf(X1[i], X1[tid]), ymin = fmaxf(Y1[i], Y1[tid]);
            float xmax = fminf(X2[i], X2[tid]), ymax = fminf(Y2[i], Y2[tid]);
            float inter = fmaxf(xmax - xmin, 0.f) * fmaxf(ymax - ymin, 0.f);
            float iou   = inter / (AR[i] + AR[tid] - inter);
            if (iou > 0.6f) SUP[tid] = 1;
        }
        __syncthreads();
    }

    if (tid < KTOP) {
        const int k = KEEP[tid];
        const int o = b * KTOP + tid;
        out[o]                 = k ? scv : 0.f;
        out[NB * KTOP + o]     = k ? (float)clsv : 0.f;
        float* bo = out + NB * KTOP * 2 + (size_t)o * 4;
        if (k) {
            bo[0] = fminf(fmaxf(rx1, 0.f), (float)(IMG_W - 1));
            bo[1] = fminf(fmaxf(ry1, 0.f), (float)(IMG_H - 1));
            bo[2] = fminf(fmaxf(rx2, 0.f), (float)(IMG_W - 1));
            bo[3] = fminf(fmaxf(ry2, 0.f), (float)(IMG_H - 1));
        } else {
            bo[0] = 0.f; bo[1] = 0.f; bo[2] = 0.f; bo[3] = 0.f;
        }
        out[NB * KTOP * 6 + o] = k ? 1.f : 0.f;
    }
}

// ---------------------------------------------------------------------------
extern "C" void kernel_launch(void* const* d_in, const int* in_sizes, int n_in,
                              void* d_out, int out_size, void* d_ws, size_t ws_size,
                              hipStream_t stream)
{
    (void)in_sizes; (void)n_in; (void)out_size; (void)ws_size;

    // setup_inputs() dict order: batch_imgs, then (cls, cnt, reg) per level
    LevelPtrs P;
    for (int i = 0; i < 5; i++) {
        P.cls[i] = (const float*)d_in[1 + 3 * i];
        P.cnt[i] = (const float*)d_in[2 + 3 * i];
        P.reg[i] = (const float*)d_in[3 + 3 * i];
    }

    const size_t BN = (size_t)NB * NLOC;
    auto alignup = [](size_t x) { return (x + 255) & ~(size_t)255; };
    char* w = (char*)d_ws;
    float* ws_scores = (float*)w;                 w += alignup(BN * sizeof(float));
    int*   ws_cls    = (int*)w;                   w += alignup(BN * sizeof(int));
    float* ws_boxes  = (float*)w;                 w += alignup(BN * 4 * sizeof(float));
    int*   top_idx   = (int*)w;                   w += alignup((size_t)NB * KTOP * sizeof(int));
    float* top_sc    = (float*)w;

    const int total  = NB * NLOC;
    const int blocks = (total + 255) / 256;
    decode_kernel<<<blocks, 256, 0, stream>>>(P, ws_scores, ws_cls, ws_boxes);
    topk_kernel<<<NB, 1024, 0, stream>>>(ws_scores, top_idx, top_sc);
    nms_kernel<<<NB, 1024, 0, stream>>>(ws_cls, ws_boxes, top_idx, top_sc,
                                        (float*)d_out);
}